// SiddonConeBP_56616258896422
// MI455X (gfx1250) — compile-verified
//
#include <hip/hip_runtime.h>

// CDNA5 / gfx1250: wave32, WMMA f32 16x16x4 used for batched ray-vector formation.
typedef __attribute__((ext_vector_type(2))) float v2f;
typedef __attribute__((ext_vector_type(8))) float v8f;

namespace {
constexpr int   kNX     = 256;   // volume voxels per axis (DX = 1.0)
constexpr int   kNU     = 128;   // detector pixels (u)
constexpr int   kNV     = 128;   // detector pixels (v)
constexpr int   kNSTEPS = 256;
constexpr float kHALF   = 128.0f;  // NX*DX/2
constexpr float kDU     = 2.0f;    // detector pitch (== DV)
}

__device__ __forceinline__ float pick3(int r, float x0, float x1, float x2) {
  return (r == 0) ? x0 : (r == 1) ? x1 : (r == 2) ? x2 : 0.0f;
}

__global__ void zero_vol_kernel(float* __restrict__ out, int n) {
  int i = blockIdx.x * blockDim.x + threadIdx.x;
  const int stride = gridDim.x * blockDim.x;
  for (; i < n; i += stride) out[i] = 0.0f;
}

// One wave32 handles: one view, one detector v-row, 16 consecutive u pixels.
// V_WMMA_F32_16X16X4_F32 computes ray = (c - s) + uu*ud + vv*vd for the 16
// rays (columns). Rows 0..2 of A are duplicated into rows 8..10 so that,
// per the 16x16 f32 D layout (lanes 0-15 hold rows 0-7, lanes 16-31 hold
// rows 8-15, same columns), lane j and lane j+16 both receive ray j's
// (rx,ry,rz). The lane pair then splits the 256-step march: lanes 0-15 do
// steps [0,128), lanes 16-31 do steps [128,256) -> full 32-lane utilization.
__global__ __launch_bounds__(256) void siddon_bp_kernel(
    const float* __restrict__ prj,  const float* __restrict__ src,
    const float* __restrict__ detc, const float* __restrict__ detu,
    const float* __restrict__ detv, float* __restrict__ vol) {
  const int lane    = threadIdx.x;                    // 0..31
  const int wave    = blockIdx.x * 8 + threadIdx.y;   // 0..16383
  const int view    = wave >> 10;                     // 16 views x 1024 waves
  const int rem     = wave & 1023;
  const int v       = rem >> 3;                       // 0..127 (detector row)
  const int ug      = rem & 7;                        // 0..7   (16-u group)
  const int j       = lane & 15;                      // ray/column index
  const int halfsel = lane >> 4;                      // which half of the march
  const int u       = ug * 16 + j;

  // Per-view geometry (wave-uniform; in fact block-uniform: 128 blocks/view).
  const float sx = src[view * 3 + 0], sy = src[view * 3 + 1], sz = src[view * 3 + 2];
  const float cxs = detc[view * 3 + 0] - sx;
  const float cys = detc[view * 3 + 1] - sy;
  const float czs = detc[view * 3 + 2] - sz;
  const float udx = detu[view * 3 + 0], udy = detu[view * 3 + 1], udz = detu[view * 3 + 2];
  const float vdx = detv[view * 3 + 0], vdy = detv[view * 3 + 1], vdz = detv[view * 3 + 2];

  const float uuu = ((float)u - 63.5f) * kDU;  // u offset for column n = lane (lanes 0-15)
  const float vvv = ((float)v - 63.5f) * kDU;  // wave-uniform v offset

  // A(16x4) f32 layout: lanes 0-15 hold {K=0,K=1} of row m=lane; lanes 16-31
  // hold {K=2,K=3} of row m=lane-16. Rows 0..2 = geometry, rows 8..10 = copy.
  const int  r  = (j < 3) ? j : ((j >= 8 && j < 11) ? (j - 8) : -1);
  const bool lo = (halfsel == 0);
  v2f a, b;
  a[0] = lo ? pick3(r, cxs, cys, czs) : pick3(r, vdx, vdy, vdz); // A[m][0] | A[m][2]
  a[1] = lo ? pick3(r, udx, udy, udz) : 0.0f;                    // A[m][1] | A[m][3]
  // B(4x16) f32 layout: lanes 0-15 hold {K=0,K=1} of column n=lane; lanes
  // 16-31 hold {K=2,K=3} of column n=lane-16.
  b[0] = lo ? 1.0f : vvv;   // B[0][n]=1  | B[2][n]=vv
  b[1] = lo ? uuu  : 0.0f;  // B[1][n]=uu | B[3][n]=0

  v8f acc = {};
  acc = __builtin_amdgcn_wmma_f32_16x16x4_f32(
      /*neg_a=*/false, a, /*neg_b=*/false, b,
      /*c_mod=*/(short)0, acc, /*reuse_a=*/false, /*reuse_b=*/false);
  // Every lane reads its ray vector from D rows 0..2 (== rows 8..10).
  const float rx = acc[0], ry = acc[1], rz = acc[2];

  // Per-ray scalar setup. Use hardware-rate v_sqrt_f32 / v_rcp_f32 (~1 ulp)
  // instead of the IEEE div/sqrt expansions (~15 VALU ops each); at this
  // geometry scale (~600 mm coords, 1 mm voxels) the error is irrelevant.
  const float n2   = fmaf(rx, rx, fmaf(ry, ry, rz * rz));
  const float L    = __builtin_amdgcn_sqrtf(n2);
  const float invL = __builtin_amdgcn_rcpf(L);
  const float dirx = rx * invL, diry = ry * invL, dirz = rz * invL;
  // Reference: where(|d| < 1e-9, 1e-9, d)
  const float dsx = (fabsf(dirx) < 1e-9f) ? 1e-9f : dirx;
  const float dsy = (fabsf(diry) < 1e-9f) ? 1e-9f : diry;
  const float dsz = (fabsf(dirz) < 1e-9f) ? 1e-9f : dirz;
  const float rdx = __builtin_amdgcn_rcpf(dsx);
  const float rdy = __builtin_amdgcn_rcpf(dsy);
  const float rdz = __builtin_amdgcn_rcpf(dsz);
  const float t1x = (-kHALF - sx) * rdx, t2x = (kHALF - sx) * rdx;
  const float t1y = (-kHALF - sy) * rdy, t2y = (kHALF - sy) * rdy;
  const float t1z = (-kHALF - sz) * rdz, t2z = (kHALF - sz) * rdz;
  float tmin = fmaxf(fmaxf(fminf(t1x, t2x), fminf(t1y, t2y)), fminf(t1z, t2z));
  float tmax = fminf(fminf(fmaxf(t1x, t2x), fmaxf(t1y, t2y)), fmaxf(t1z, t2z));
  tmin = fminf(fmaxf(tmin, 0.0f), L);
  tmax = fminf(fmaxf(tmax, 0.0f), L);
  const float step = fmaxf(tmax - tmin, 0.0f) * (1.0f / (float)kNSTEPS);

  const float pval = prj[(view * kNV + v) * kNU + u];
  const float w    = pval * step;

  if (step > 0.0f) {
    // Strength-reduced sample position (voxel units, DX = 1):
    //   p(fi) = (s + HALF + tmin*dir) + fi * (step*dir)
    const float bx = fmaf(tmin, dirx, sx + kHALF);
    const float by = fmaf(tmin, diry, sy + kHALF);
    const float bz = fmaf(tmin, dirz, sz + kHALF);
    const float ix_ = step * dirx, iy_ = step * diry, iz_ = step * dirz;
    float fi = (float)(halfsel * (kNSTEPS / 2)) + 0.5f;
#pragma unroll 8
    for (int k = 0; k < kNSTEPS / 2; ++k) {
      const float px = fmaf(fi, ix_, bx);
      const float py = fmaf(fi, iy_, by);
      const float pz = fmaf(fi, iz_, bz);
      const int ix = (int)floorf(px);
      const int iy = (int)floorf(py);
      const int iz = (int)floorf(pz);
      fi += 1.0f;
      if ((unsigned)ix < (unsigned)kNX && (unsigned)iy < (unsigned)kNX &&
          (unsigned)iz < (unsigned)kNX) {
        // Non-returning f32 atomic add: volume (67 MB) is L2-resident (192 MB
        // L2), so the RMW stays on-chip; result discarded -> STOREcnt path.
        (void)__hip_atomic_fetch_add(&vol[((iz * kNX) + iy) * kNX + ix], w,
                                     __ATOMIC_RELAXED, __HIP_MEMORY_SCOPE_AGENT);
      }
    }
  }
}

extern "C" void kernel_launch(void* const* d_in, const int* in_sizes, int n_in,
                              void* d_out, int out_size, void* d_ws, size_t ws_size,
                              hipStream_t stream) {
  (void)in_sizes; (void)n_in; (void)d_ws; (void)ws_size;
  const float* prj  = (const float*)d_in[0];
  const float* src  = (const float*)d_in[1];
  const float* detc = (const float*)d_in[2];
  const float* detu = (const float*)d_in[3];
  const float* detv = (const float*)d_in[4];
  float* vol = (float*)d_out;

  // d_out is poisoned by the harness; zero it every call (deterministic).
  zero_vol_kernel<<<2048, 256, 0, stream>>>(vol, out_size);

  // 16 views x 128 v-rows x 8 u-groups = 16384 waves = 2048 blocks x 8 waves,
  // every wave fully populated (WMMA requires EXEC all-ones).
  dim3 blk(32, 8, 1);
  siddon_bp_kernel<<<2048, blk, 0, stream>>>(prj, src, detc, detu, detv, vol);
}